// CheapBiMamba3D_18777597018414
// MI455X (gfx1250) — compile-verified
//
#include <hip/hip_runtime.h>
#include <math.h>

typedef float v2f __attribute__((ext_vector_type(2)));
typedef float v4f __attribute__((ext_vector_type(4)));
typedef float v8f __attribute__((ext_vector_type(8)));

#define HAS_WMMA_F32X4 __has_builtin(__builtin_amdgcn_wmma_f32_16x16x4_f32)

namespace {
constexpr int Cch  = 256;           // channels
constexpr int Dd   = 16;            // depth
constexpr int Hh   = 128, Ww = 128;
constexpr int CR   = 32;            // reduced channels
constexpr int DST  = 16;            // state dim
constexpr int DI   = 64;            // inner dim (EXPAND*CR)
constexpr int DTR  = 2;             // dt rank
constexpr int Lseq = 1024;          // Hs*Ws tokens per sequence
constexpr int NSEQ = 16;            // = Dd sequences
constexpr int T    = NSEQ * Lseq;   // 16384 tokens
constexpr int PLANE = Dd * Hh * Ww; // per-channel plane = 262144
}

__device__ __forceinline__ float silu_f(float v) {
  return v * (1.0f / (1.0f + __expf(-v)));
}

// ---------------------------------------------------------------------------
// K1: input projection at strided positions: tokp[t][r] = sum_c x[c,d,4hs,4ws]*w_in[r,c]
// One wave -> 16-token tile x 32 r outputs, fp32 WMMA 16x16x4 over K=256.
// x loads are non-temporal: each strided line is used once, keep L2 clean.
// ---------------------------------------------------------------------------
#if HAS_WMMA_F32X4
__global__ void k_proj_in(const float* __restrict__ x, const float* __restrict__ w_in,
                          float* __restrict__ tokp) {
  const int lane = threadIdx.x & 31;
  const int tile = blockIdx.x * 8 + (threadIdx.x >> 5);   // 0..1023
  const int t0 = tile * 16;
  const int m = lane & 15;
  const int half = lane >> 4;
  const int t = t0 + m;
  const int d = t >> 10, hs = (t >> 5) & 31, wsx = t & 31;
  const int base = d * (Hh * Ww) + (hs * 4) * Ww + wsx * 4;   // offset in channel plane
  v8f acc0 = {}; v8f acc1 = {};
  for (int k0 = 0; k0 < Cch; k0 += 4) {
    const int ck = k0 + half * 2;
    v2f a;
    a.x = __builtin_nontemporal_load(x + ck * PLANE + base);
    a.y = __builtin_nontemporal_load(x + (ck + 1) * PLANE + base);
    v2f b0 = *(const v2f*)(w_in + m * Cch + ck);            // r = m
    v2f b1 = *(const v2f*)(w_in + (16 + m) * Cch + ck);     // r = 16+m
    acc0 = __builtin_amdgcn_wmma_f32_16x16x4_f32(false, a, false, b0, (short)0, acc0, false, false);
    acc1 = __builtin_amdgcn_wmma_f32_16x16x4_f32(false, a, false, b1, (short)0, acc1, false, false);
  }
#pragma unroll
  for (int i = 0; i < 8; ++i) {
    const int tr = t0 + i + half * 8;
    tokp[tr * CR + m]      = acc0[i];
    tokp[tr * CR + 16 + m] = acc1[i];
  }
}
#else
__global__ void k_proj_in(const float* __restrict__ x, const float* __restrict__ w_in,
                          float* __restrict__ tokp) {
  const int r = threadIdx.x & 31;
  const int t = blockIdx.x * 8 + (threadIdx.x >> 5);
  const int d = t >> 10, hs = (t >> 5) & 31, wsx = t & 31;
  const int base = d * (Hh * Ww) + (hs * 4) * Ww + wsx * 4;
  float acc = 0.f;
  for (int c = 0; c < Cch; ++c) acc += x[c * PLANE + base] * w_in[r * Cch + c];
  tokp[t * CR + r] = acc;
}
#endif

// ---------------------------------------------------------------------------
// K2: LayerNorm over CR=32 per token (one wave per token, in place).
// ---------------------------------------------------------------------------
__global__ void k_ln(float* __restrict__ tokp, const float* __restrict__ ln_w,
                     const float* __restrict__ ln_b) {
  const int r = threadIdx.x & 31;
  const int t = blockIdx.x * 8 + (threadIdx.x >> 5);
  float v = tokp[t * CR + r];
  float s = v;
  for (int off = 16; off >= 1; off >>= 1) s += __shfl_xor(s, off, 32);
  const float mu = s * (1.0f / 32.0f);
  const float dv = v - mu;
  float s2 = dv * dv;
  for (int off = 16; off >= 1; off >>= 1) s2 += __shfl_xor(s2, off, 32);
  const float var = s2 * (1.0f / 32.0f);
  tokp[t * CR + r] = dv * rsqrtf(var + 1e-5f) * ln_w[r] + ln_b[r];
}

// ---------------------------------------------------------------------------
// K3: xz = tokn @ in_w^T  (split into x and z halves). 2 tokens per block.
// ---------------------------------------------------------------------------
__global__ void k_xz(const float* __restrict__ tokn, const float* __restrict__ in_w,
                     float* __restrict__ xb, float* __restrict__ zb) {
  __shared__ float wT[CR * 128];   // transposed weights [r][e]
  __shared__ float st[2][CR];
  const int tid = threadIdx.x;
  for (int i = tid; i < CR * 128; i += 256) {
    const int e = i >> 5, r = i & 31;
    wT[r * 128 + e] = in_w[e * CR + r];
  }
  if (tid < 64) {
    const int tt = tid >> 5, r = tid & 31;
    st[tt][r] = tokn[(blockIdx.x * 2 + tt) * CR + r];
  }
  __syncthreads();
  const int tloc = tid >> 7, e = tid & 127;
  const int t = blockIdx.x * 2 + tloc;
  float acc = 0.f;
#pragma unroll 8
  for (int r = 0; r < CR; ++r) acc += st[tloc][r] * wT[r * 128 + e];
  if (e < DI) xb[t * DI + e] = acc;
  else        zb[t * DI + (e - DI)] = acc;
}

// ---------------------------------------------------------------------------
// K4: depthwise causal conv (width 4) + silu. dir=1 processes flipped sequence.
// ---------------------------------------------------------------------------
__global__ void k_conv(const float* __restrict__ xb, const float* __restrict__ cw,
                       const float* __restrict__ cb, float* __restrict__ xs, int dir) {
  const int idx = blockIdx.x * 256 + threadIdx.x;   // over T*DI
  const int ch = idx & 63;
  const int t  = idx >> 6;
  const int n = t >> 10, l = t & 1023;
  float acc = cb[ch];
#pragma unroll
  for (int k = 0; k < 4; ++k) {
    const int ll = dir ? (l + 3 - k) : (l - 3 + k);
    if (ll >= 0 && ll < Lseq)
      acc += cw[ch * 4 + k] * xb[((n << 10) + ll) * DI + ch];
  }
  xs[idx] = silu_f(acc);
}

// ---------------------------------------------------------------------------
// K5: dbc = xs @ xproj_w^T (34 outputs: 2 dt-rank, 16 B, 16 C). 4 tokens/block.
// ---------------------------------------------------------------------------
__global__ void k_xproj(const float* __restrict__ xs, const float* __restrict__ xw,
                        float* __restrict__ dtraw, float* __restrict__ Bm,
                        float* __restrict__ Cm) {
  __shared__ float sx[4][DI];
  __shared__ float wpT[DI * 34];
  const int tid = threadIdx.x;
  for (int i = tid; i < 34 * DI; i += 256) {
    const int j = i >> 6, ch = i & 63;
    wpT[ch * 34 + j] = xw[i];
  }
  const int tloc = tid >> 6, j = tid & 63;
  const int t = blockIdx.x * 4 + tloc;
  sx[tloc][j] = xs[t * DI + j];
  __syncthreads();
  if (j < 34) {
    float acc = 0.f;
#pragma unroll 8
    for (int ch = 0; ch < DI; ++ch) acc += sx[tloc][ch] * wpT[ch * 34 + j];
    if (j < DTR)            dtraw[t * DTR + j] = acc;
    else if (j < DTR + DST) Bm[t * DST + (j - DTR)] = acc;
    else                    Cm[t * DST + (j - DTR - DST)] = acc;
  }
}

// ---------------------------------------------------------------------------
// K6: dt = softplus(dtraw @ dt_w^T + dt_b)
// ---------------------------------------------------------------------------
__global__ void k_dt(const float* __restrict__ dtraw, const float* __restrict__ dtw,
                     const float* __restrict__ dtbias, float* __restrict__ dtb) {
  const int idx = blockIdx.x * 256 + threadIdx.x;   // over T*DI
  const int ch = idx & 63;
  const int t  = idx >> 6;
  const float v = dtraw[t * 2] * dtw[ch * 2] + dtraw[t * 2 + 1] * dtw[ch * 2 + 1] + dtbias[ch];
  dtb[idx] = (v > 20.0f) ? v : log1pf(__expf(v));
}

// ---------------------------------------------------------------------------
// K7: selective scan. One wave = one sequence x 2 channels; lane = s + 16*dlocal.
// Fuses dA/dBx compute, state reduce over s (shfl_xor width 16), +D*x, silu(z)
// gating, and the y write.
// ---------------------------------------------------------------------------
__global__ void k_scan(const float* __restrict__ dtb, const float* __restrict__ xs,
                       const float* __restrict__ Bm, const float* __restrict__ Cm,
                       const float* __restrict__ zb, const float* __restrict__ A_log,
                       const float* __restrict__ Dv, float* __restrict__ yb, int dir) {
  const int lane = threadIdx.x & 31;
  const int gid = blockIdx.x * 8 + (threadIdx.x >> 5);  // 0..511
  const int n = gid >> 5;
  const int ch = (gid & 31) * 2 + (lane >> 4);
  const int s = lane & 15;
  const float a = -__expf(A_log[ch * DST + s]);
  const float Dch = Dv[ch];
  float h = 0.f;
  for (int step = 0; step < Lseq; ++step) {
    const int l = dir ? (Lseq - 1 - step) : step;
    const int tt = (n << 10) + l;
    const float dtc = dtb[tt * DI + ch];
    const float xv  = xs[tt * DI + ch];
    const float bv  = Bm[tt * DST + s];
    const float cv  = Cm[tt * DST + s];
    h = __expf(dtc * a) * h + dtc * xv * bv;
    float p = h * cv;
    p += __shfl_xor(p, 1, 16);
    p += __shfl_xor(p, 2, 16);
    p += __shfl_xor(p, 4, 16);
    p += __shfl_xor(p, 8, 16);
    if (s == 0) {
      const float zv = zb[tt * DI + ch];
      yb[tt * DI + ch] = (p + Dch * xv) * silu_f(zv);
    }
  }
}

// ---------------------------------------------------------------------------
// K8: ym (+)= 0.5 * (y @ out_w^T). 8 tokens per block.
// ---------------------------------------------------------------------------
__global__ void k_ymacc(const float* __restrict__ yb, const float* __restrict__ ow,
                        float* __restrict__ ym, int dir) {
  __shared__ float owT[DI * CR];
  __shared__ float sy[8][DI];
  const int tid = threadIdx.x;
  for (int i = tid; i < DI * CR; i += 256) {
    const int r = i >> 6, ch = i & 63;
    owT[ch * CR + r] = ow[i];
  }
  for (int i = tid; i < 8 * DI; i += 256)
    sy[i >> 6][i & 63] = yb[(blockIdx.x * 8 + (i >> 6)) * DI + (i & 63)];
  __syncthreads();
  const int tloc = tid >> 5, r = tid & 31;
  const int t = blockIdx.x * 8 + tloc;
  float acc = 0.f;
#pragma unroll 8
  for (int ch = 0; ch < DI; ++ch) acc += sy[tloc][ch] * owT[ch * CR + r];
  acc *= 0.5f;
  if (dir == 0) ym[t * CR + r] = acc;
  else          ym[t * CR + r] += acc;
}

// ---------------------------------------------------------------------------
// K9: P[t][c] = sum_r ym[t][r] * w_out[c][r].  fp32 WMMA, K=32 (8 steps).
// Block = 8 waves; wave w covers channel tiles {w, w+8} of one 16-token tile.
// P stores stay regular-temporal: P (16.8 MB) is re-read 4x by k_up and we
// want it resident in the 192 MB L2.
// ---------------------------------------------------------------------------
#if HAS_WMMA_F32X4
__global__ void k_proj_out(const float* __restrict__ ym, const float* __restrict__ w_out,
                           float* __restrict__ P) {
  const int lane = threadIdx.x & 31;
  const int wv = threadIdx.x >> 5;
  const int t0 = blockIdx.x * 16;
  const int m = lane & 15;
  const int half = lane >> 4;
#pragma unroll
  for (int cti = 0; cti < 2; ++cti) {
    const int c0 = (wv + cti * 8) * 16;
    v8f acc = {};
    for (int k0 = 0; k0 < CR; k0 += 4) {
      v2f a = *(const v2f*)(ym + (t0 + m) * CR + k0 + half * 2);
      v2f b = *(const v2f*)(w_out + (c0 + m) * CR + k0 + half * 2);
      acc = __builtin_amdgcn_wmma_f32_16x16x4_f32(false, a, false, b, (short)0, acc, false, false);
    }
#pragma unroll
    for (int i = 0; i < 8; ++i)
      P[(t0 + i + half * 8) * Cch + c0 + m] = acc[i];
  }
}
#else
__global__ void k_proj_out(const float* __restrict__ ym, const float* __restrict__ w_out,
                           float* __restrict__ P) {
  const int idx = blockIdx.x * 256 + threadIdx.x;   // over T*Cch
  const int c = idx & 255;
  const int t = idx >> 8;
  float acc = 0.f;
#pragma unroll 8
  for (int r = 0; r < CR; ++r) acc += ym[t * CR + r] * w_out[c * CR + r];
  P[idx] = acc;
}
#endif

// ---------------------------------------------------------------------------
// K10: nearest 4x4 upsample broadcast store: out[c,d,h,w] = P[t(d,h/4,w/4)][c].
// float4 non-temporal stores (write-once 268 MB stream; don't churn L2),
// fully coalesced; this is the bandwidth floor of the whole op.
// ---------------------------------------------------------------------------
__global__ void k_up(const float* __restrict__ P, float* __restrict__ out) {
  const unsigned idx = blockIdx.x * 256u + threadIdx.x;  // 2^24 float4 stores
  const int wq = idx & 31;
  const int h  = (idx >> 5) & 127;
  const int d  = (idx >> 12) & 15;
  const int c  = idx >> 16;
  const int t  = (d << 10) + ((h >> 2) << 5) + wq;
  const float v = P[(size_t)t * Cch + c];
  v4f val = {v, v, v, v};
  v4f* dst = (v4f*)(out + (((size_t)(c * 16 + d) * 128 + h) << 7) + (wq << 2));
  __builtin_nontemporal_store(val, dst);
}

// ---------------------------------------------------------------------------
extern "C" void kernel_launch(void* const* d_in, const int* in_sizes, int n_in,
                              void* d_out, int out_size, void* d_ws, size_t ws_size,
                              hipStream_t stream) {
  (void)in_sizes; (void)n_in; (void)out_size; (void)ws_size;
  const float* x     = (const float*)d_in[0];
  const float* w_in  = (const float*)d_in[1];
  const float* w_out = (const float*)d_in[2];
  const float* ln_w  = (const float*)d_in[3];
  const float* ln_b  = (const float*)d_in[4];
  // per direction: in_w, conv_w, conv_b, xproj_w, dt_w, dt_b, A_log, D, out_w
  const float* mw[2][9];
  for (int dir = 0; dir < 2; ++dir)
    for (int j = 0; j < 9; ++j)
      mw[dir][j] = (const float*)d_in[5 + dir * 9 + j];
  float* out = (float*)d_out;

  float* ws = (float*)d_ws;
  size_t o = 0;
  float* tokn  = ws + o; o += (size_t)T * CR;
  float* ym    = ws + o; o += (size_t)T * CR;
  float* Pbuf  = ws + o; o += (size_t)T * Cch;
  float* xb    = ws + o; o += (size_t)T * DI;
  float* zb    = ws + o; o += (size_t)T * DI;
  float* xsb   = ws + o; o += (size_t)T * DI;
  float* dtraw = ws + o; o += (size_t)T * DTR;
  float* Bmb   = ws + o; o += (size_t)T * DST;
  float* Cmb   = ws + o; o += (size_t)T * DST;
  float* dtb   = ws + o; o += (size_t)T * DI;
  float* ybuf  = ws + o; o += (size_t)T * DI;

  hipLaunchKernelGGL(k_proj_in, dim3(T / 16 / 8), dim3(256), 0, stream, x, w_in, tokn);
  hipLaunchKernelGGL(k_ln,      dim3(T / 8),      dim3(256), 0, stream, tokn, ln_w, ln_b);

  for (int dir = 0; dir < 2; ++dir) {
    hipLaunchKernelGGL(k_xz,    dim3(T / 2),           dim3(256), 0, stream,
                       tokn, mw[dir][0], xb, zb);
    hipLaunchKernelGGL(k_conv,  dim3(T * DI / 256),    dim3(256), 0, stream,
                       xb, mw[dir][1], mw[dir][2], xsb, dir);
    hipLaunchKernelGGL(k_xproj, dim3(T / 4),           dim3(256), 0, stream,
                       xsb, mw[dir][3], dtraw, Bmb, Cmb);
    hipLaunchKernelGGL(k_dt,    dim3(T * DI / 256),    dim3(256), 0, stream,
                       dtraw, mw[dir][4], mw[dir][5], dtb);
    hipLaunchKernelGGL(k_scan,  dim3(64),              dim3(256), 0, stream,
                       dtb, xsb, Bmb, Cmb, zb, mw[dir][6], mw[dir][7], ybuf, dir);
    hipLaunchKernelGGL(k_ymacc, dim3(T / 8),           dim3(256), 0, stream,
                       ybuf, mw[dir][8], ym, dir);
  }

  hipLaunchKernelGGL(k_proj_out, dim3(T / 16), dim3(256), 0, stream, ym, w_out, Pbuf);
  hipLaunchKernelGGL(k_up, dim3((unsigned)((size_t)Cch * Dd * Hh * Ww / 4 / 256)),
                     dim3(256), 0, stream, Pbuf, out);
}